// MatrixFactorizationModel_80960133530116
// MI455X (gfx1250) — compile-verified
//
#include <hip/hip_runtime.h>

typedef __attribute__((ext_vector_type(2))) float v2f;
typedef __attribute__((ext_vector_type(8))) float v8f;

#define FEAT 50
#define DIM  128
#define TILE 16
// pad LDS rows by 4 floats: column reads stride 132 -> bank stride 4 -> conflict-free
#define DP   (DIM + 4)

__global__ __launch_bounds__(256)
void mf_predict_kernel(const int*   __restrict__ user_ids,
                       const int*   __restrict__ item_ids,
                       const int*   __restrict__ ufi,
                       const float* __restrict__ ufv,
                       const int*   __restrict__ ifi,
                       const float* __restrict__ ifv,
                       const float* __restrict__ user_emb,
                       const float* __restrict__ item_emb,
                       const float* __restrict__ user_feat_emb,
                       const float* __restrict__ item_feat_emb,
                       float*       __restrict__ out,
                       int nB)
{
    __shared__ float su[TILE][DP];
    __shared__ float sv[TILE][DP];
    __shared__ float sc[TILE][TILE];

    const int tid     = threadIdx.x;
    const int s_local = tid >> 4;      // 0..15: sample within tile
    const int lane16  = tid & 15;      // 0..15: dim-chunk owner
    const int dim0    = lane16 * 8;    // 8 dims per thread

    int b = blockIdx.x * TILE + s_local;
    if (b >= nB) b = nB - 1;           // clamp (nB is a multiple of 16 anyway)

    // ---- Phase 1: gather + weighted accumulate (fp32, float4 vector loads) ----
    const float* ubase = user_emb + (size_t)user_ids[b] * DIM + dim0;
    const float* vbase = item_emb + (size_t)item_ids[b] * DIM + dim0;
    float4 au0 = *(const float4*)(ubase);
    float4 au1 = *(const float4*)(ubase + 4);
    float4 av0 = *(const float4*)(vbase);
    float4 av1 = *(const float4*)(vbase + 4);

    const int*   uidx = ufi + (size_t)b * FEAT;
    const float* uval = ufv + (size_t)b * FEAT;
    const int*   iidx = ifi + (size_t)b * FEAT;
    const float* ival = ifv + (size_t)b * FEAT;

    #pragma unroll 2
    for (int f = 0; f < FEAT; ++f) {
        const int   ui = uidx[f];
        const float uw = uval[f];
        const int   ii = iidx[f];
        const float iw = ival[f];

        const float* ue = user_feat_emb + (size_t)ui * DIM + dim0;
        const float* ie = item_feat_emb + (size_t)ii * DIM + dim0;
        float4 e0 = *(const float4*)(ue);
        float4 e1 = *(const float4*)(ue + 4);
        float4 g0 = *(const float4*)(ie);
        float4 g1 = *(const float4*)(ie + 4);

        au0.x += uw * e0.x; au0.y += uw * e0.y; au0.z += uw * e0.z; au0.w += uw * e0.w;
        au1.x += uw * e1.x; au1.y += uw * e1.y; au1.z += uw * e1.z; au1.w += uw * e1.w;
        av0.x += iw * g0.x; av0.y += iw * g0.y; av0.z += iw * g0.z; av0.w += iw * g0.w;
        av1.x += iw * g1.x; av1.y += iw * g1.y; av1.z += iw * g1.z; av1.w += iw * g1.w;
    }

    // stage u,v tiles in LDS (16 samples x 128 dims)
    *(float4*)&su[s_local][dim0]     = au0;
    *(float4*)&su[s_local][dim0 + 4] = au1;
    *(float4*)&sv[s_local][dim0]     = av0;
    *(float4*)&sv[s_local][dim0 + 4] = av1;
    __syncthreads();

    // ---- Phase 2: dot products via V_WMMA_F32_16X16X4_F32 (wave 0 only; EXEC all-1s) ----
    if (tid < 32) {
        const int m  = tid & 15;                 // row (A) / col (B) index
        const int kb = (tid < 16) ? 0 : 2;       // K pair base per half-wave (ISA layout)
        v8f c = {0.f, 0.f, 0.f, 0.f, 0.f, 0.f, 0.f, 0.f};

        #pragma unroll
        for (int k0 = 0; k0 < DIM; k0 += 4) {
            v2f a, bm;
            a.x  = su[m][k0 + kb];
            a.y  = su[m][k0 + kb + 1];
            bm.x = sv[m][k0 + kb];      // B[k][n] = v[n][k]  (B = V^T)
            bm.y = sv[m][k0 + kb + 1];
            c = __builtin_amdgcn_wmma_f32_16x16x4_f32(
                    /*neg_a=*/false, a, /*neg_b=*/false, bm,
                    /*c_mod=*/(short)0, c, /*reuse_a=*/false, /*reuse_b=*/false);
        }

        // scatter C (16x16) to LDS: VGPR r holds M = r (+8 for upper half-wave), N = lane&15
        const int mbase = (tid < 16) ? 0 : 8;
        #pragma unroll
        for (int r = 0; r < 8; ++r) {
            sc[mbase + r][m] = c[r];
        }
    }
    __syncthreads();

    // ---- Phase 3: emit diagonal = per-sample dot products ----
    if (tid < TILE) {
        const int ob = blockIdx.x * TILE + tid;
        if (ob < nB) out[ob] = sc[tid][tid];
    }
}

extern "C" void kernel_launch(void* const* d_in, const int* in_sizes, int n_in,
                              void* d_out, int out_size, void* d_ws, size_t ws_size,
                              hipStream_t stream) {
    const int*   user_ids      = (const int*)  d_in[0];
    const int*   item_ids      = (const int*)  d_in[1];
    const int*   ufi           = (const int*)  d_in[2];
    const float* ufv           = (const float*)d_in[3];
    const int*   ifi           = (const int*)  d_in[4];
    const float* ifv           = (const float*)d_in[5];
    const float* user_emb      = (const float*)d_in[6];
    const float* item_emb      = (const float*)d_in[7];
    const float* user_feat_emb = (const float*)d_in[8];
    const float* item_feat_emb = (const float*)d_in[9];
    float*       out           = (float*)d_out;

    const int nB = in_sizes[0];               // 16384
    const int blocks = (nB + TILE - 1) / TILE;

    mf_predict_kernel<<<blocks, 256, 0, stream>>>(
        user_ids, item_ids, ufi, ufv, ifi, ifv,
        user_emb, item_emb, user_feat_emb, item_feat_emb,
        out, nB);
}